// SelfAttention_16836271800502
// MI455X (gfx1250) — compile-verified
//
#include <hip/hip_runtime.h>

#define C_DIM 64
#define N_DIM 4096
#define B_DIM 4
#define SLDA  132   // padded LDS row stride (floats) for the 64x128 score tile
#define QLDA  136   // padded LDS row stride for the staged Q chunk (16B aligned,
                    // 2*136 % 64 == 16 -> hi/lo lane halves use disjoint banks)

#if defined(__has_builtin)
#if __has_builtin(__builtin_amdgcn_global_load_async_to_lds_b128) && \
    __has_builtin(__builtin_amdgcn_s_wait_asynccnt)
#define HAVE_ASYNC_LDS 1
#endif
#endif

typedef float v2f __attribute__((ext_vector_type(2)));
typedef float v8f __attribute__((ext_vector_type(8)));

// 16-byte packet type for the async global->LDS copy, per the builtin's
// signature (pointer to <4 x i32>).
typedef int v4i __attribute__((vector_size(16)));
typedef __attribute__((address_space(1))) v4i* g128_t;  // global 16B packet
typedef __attribute__((address_space(3))) v4i* l128_t;  // LDS    16B packet

__device__ __forceinline__ v8f wmma_f32_16x16x4(v2f a, v2f b, v8f c) {
  // D = A(16x4 f32) * B(4x16 f32) + C(16x16 f32)
  return __builtin_amdgcn_wmma_f32_16x16x4_f32(false, a, false, b, (short)0, c,
                                               false, false);
}

// ---------------------------------------------------------------------------
// Kernel 1: K/Q/V projection.  k|q|v = W{k,q,v} (64x64) @ xf (64x4096) per batch.
// Block = 256 threads (8 waves), computes a 64 x 128 output tile of all three.
// ---------------------------------------------------------------------------
__global__ __launch_bounds__(256) void qkv_proj_kernel(
    const float* __restrict__ x, const float* __restrict__ Wk,
    const float* __restrict__ Wq, const float* __restrict__ Wv,
    float* __restrict__ kb, float* __restrict__ qb, float* __restrict__ vb) {
  const int b    = blockIdx.y;
  const int n0   = blockIdx.x * 128;
  const int tid  = threadIdx.x;
  const int wave = tid >> 5, lane = tid & 31;
  const int hi   = lane >> 4, lo = lane & 15;

  const size_t base = (size_t)b * C_DIM * N_DIM;
  const float* xf = x + base;
  const int o_off = (wave & 3) * 16;       // output-channel subtile
  const int colb  = (wave >> 2) * 16;      // column-half within the j*32 stripe

  for (int j = 0; j < 4; ++j) {
    const int col = n0 + colb + j * 32 + lo;
    // B fragments: B[K=c][N=col] = xf[c, col]; lanes 0-15 carry K=c0,c0+1,
    // lanes 16-31 carry K=c0+2,c0+3 (per 16x16x4 layout).
    v2f bf[16];
#pragma unroll
    for (int kk = 0; kk < 16; ++kk) {
      const int c0 = kk * 4 + 2 * hi;
      bf[kk].x = xf[(size_t)c0 * N_DIM + col];
      bf[kk].y = xf[(size_t)(c0 + 1) * N_DIM + col];
    }
#pragma unroll
    for (int mat = 0; mat < 3; ++mat) {
      const float* Wm = (mat == 0) ? Wk : (mat == 1) ? Wq : Wv;
      v8f acc = {};
#pragma unroll
      for (int kk = 0; kk < 16; ++kk) {
        const int c0 = kk * 4 + 2 * hi;
        v2f a;  // A[M=o][K=c] = W[o,c]
        a.x = Wm[(o_off + lo) * 64 + c0];
        a.y = Wm[(o_off + lo) * 64 + c0 + 1];
        acc = wmma_f32_16x16x4(a, bf[kk], acc);
      }
      float* dst = ((mat == 0) ? kb : (mat == 1) ? qb : vb) + base;
#pragma unroll
      for (int r = 0; r < 8; ++r)  // D: VGPR r -> row r (lanes 0-15) / r+8
        dst[(size_t)(o_off + r + 8 * hi) * N_DIM + col] = acc[r];
    }
  }
}

// ---------------------------------------------------------------------------
// Kernel 2: fused flash-attention + W2 + BN + ReLU + residual.
// Block owns 64 "n" rows of the score matrix; loops m in chunks of 128.
// Q chunk is staged to LDS via the CDNA5 async global->LDS path (ASYNCcnt).
// ---------------------------------------------------------------------------
__global__ __launch_bounds__(256) void attn_fused_kernel(
    const float* __restrict__ kb, const float* __restrict__ qb,
    const float* __restrict__ vb, const float* __restrict__ W2,
    const float* __restrict__ gamma, const float* __restrict__ beta,
    const float* __restrict__ rmean, const float* __restrict__ rvar,
    const float* __restrict__ x, float* __restrict__ out) {
  __shared__ float sS[64 * SLDA];    // score / prob tile (64 n x 128 m), padded
  __shared__ float sQ[64 * QLDA];    // staged Q chunk (64 c x 128 m), padded
  __shared__ float sAtt[64 * 64];    // att tile (64 c x 64 n)
  __shared__ float sRowMax[64], sRowSum[64], sCorr[64], sInv[64], sBias[64];

  const int b    = blockIdx.y;
  const int nblk = blockIdx.x * 64;
  const int tid  = threadIdx.x;
  const int wave = tid >> 5, lane = tid & 31;
  const int hi   = lane >> 4, lo = lane & 15;

  const size_t base = (size_t)b * C_DIM * N_DIM;
  const float* kp = kb + base;
  const float* qp = qb + base;
  const float* vp = vb + base;

  if (tid < 64) {
    sRowMax[tid] = -3.0e38f;
    sRowSum[tid] = 0.0f;
    const float inv = gamma[tid] * rsqrtf(rvar[tid] + 1e-5f);
    sInv[tid]  = inv;
    sBias[tid] = beta[tid] - rmean[tid] * inv;
  }
  __syncthreads();

  // GEMM1 (S = K^T Q) mapping: wave -> n-subtile + m-half of the 128 chunk.
  const int nsub1 = wave & 3;
  const int mhalf = wave >> 2;
  // A fragment A[n][c] = k[c, n] is chunk-invariant: hoist all 16 K-steps.
  v2f afragK[16];
  {
    const int ncol = nblk + nsub1 * 16 + lo;
#pragma unroll
    for (int kk = 0; kk < 16; ++kk) {
      const int c0 = kk * 4 + 2 * hi;
      afragK[kk].x = kp[(size_t)c0 * N_DIM + ncol];
      afragK[kk].y = kp[(size_t)(c0 + 1) * N_DIM + ncol];
    }
  }

  // GEMM2 (O = V P~^T) mapping: wave -> c-subtile, two n-subtiles.
  const int c_off = (wave & 3) * 16;
  const int n2a   = (wave >> 2) * 2;
  const int n2b   = n2a + 1;
  v8f acc0 = {}, acc1 = {};

  for (int ch = 0; ch < 32; ++ch) {
    const int m0 = ch * 128;

    // ---- stage the 64x128 Q chunk into LDS (2048 x 16B, 8 per thread) ----
#if HAVE_ASYNC_LDS
#pragma unroll
    for (int i = 0; i < 8; ++i) {
      const int id   = tid + i * 256;      // 0..2047
      const int row  = id >> 5;            // c: 0..63
      const int colc = (id & 31) << 2;     // m within chunk: 0..124 step 4
      __builtin_amdgcn_global_load_async_to_lds_b128(
          (g128_t)(qp + (size_t)row * N_DIM + m0 + colc),
          (l128_t)(&sQ[row * QLDA + colc]),
          0, 0);
    }
    __builtin_amdgcn_s_wait_asynccnt(0);
    __syncthreads();
#else
#pragma unroll
    for (int i = 0; i < 8; ++i) {
      const int id   = tid + i * 256;
      const int row  = id >> 5;
      const int colc = (id & 31) << 2;
      const float4 d =
          *(const float4*)(qp + (size_t)row * N_DIM + m0 + colc);
      *(float4*)(&sQ[row * QLDA + colc]) = d;
    }
    __syncthreads();
#endif

    // ---- GEMM1: this wave's 4 score subtiles of the 64x128 chunk ----
    for (int j = 0; j < 4; ++j) {
      const int mc = mhalf * 64 + j * 16 + lo;
      v8f s = {};
#pragma unroll
      for (int kk = 0; kk < 16; ++kk) {
        const int c0 = kk * 4 + 2 * hi;
        v2f bf;  // B[K=c][N=m] = q[c, m] from LDS
        bf.x = sQ[c0 * QLDA + mc];
        bf.y = sQ[(c0 + 1) * QLDA + mc];
        s = wmma_f32_16x16x4(afragK[kk], bf, s);
      }
#pragma unroll
      for (int r = 0; r < 8; ++r)
        sS[(nsub1 * 16 + r + 8 * hi) * SLDA + mc] = s[r];
    }
    __syncthreads();

    // ---- online softmax over the 64 rows (4 threads per row) ----
    {
      const int row  = tid >> 2;
      const int part = tid & 3;
      float* srow = sS + row * SLDA + part * 32;
      float mx = -3.0e38f;
#pragma unroll
      for (int i = 0; i < 32; ++i) mx = fmaxf(mx, srow[i]);
      mx = fmaxf(mx, __shfl_xor(mx, 1, 32));
      mx = fmaxf(mx, __shfl_xor(mx, 2, 32));
      const float oldm = sRowMax[row];
      const float newm = fmaxf(oldm, mx);
      const float corr = __expf(oldm - newm);
      float lsum = 0.0f;
#pragma unroll
      for (int i = 0; i < 32; ++i) {
        const float e = __expf(srow[i] - newm);
        srow[i] = e;
        lsum += e;
      }
      lsum += __shfl_xor(lsum, 1, 32);
      lsum += __shfl_xor(lsum, 2, 32);
      if (part == 0) {
        sRowMax[row] = newm;
        sCorr[row]   = corr;
        sRowSum[row] = sRowSum[row] * corr + lsum;
      }
    }
    __syncthreads();

    // ---- GEMM2: rescale accumulators (per-lane n column) + accumulate ----
    {
      const float sc0 = sCorr[n2a * 16 + lo];
      const float sc1 = sCorr[n2b * 16 + lo];
#pragma unroll
      for (int r = 0; r < 8; ++r) { acc0[r] *= sc0; acc1[r] *= sc1; }
#pragma unroll 8
      for (int ks = 0; ks < 32; ++ks) {
        const int mloc = ks * 4 + 2 * hi;
        // A[M=c][K=m] = v[c, m]: two consecutive floats per lane.
        v2f a = *(const v2f*)(vp + (size_t)(c_off + lo) * N_DIM + m0 + mloc);
        // B[K=m][N=n] = P~[n, m]: row of sS, two consecutive floats.
        v2f b0 = *(const v2f*)(sS + (n2a * 16 + lo) * SLDA + mloc);
        v2f b1 = *(const v2f*)(sS + (n2b * 16 + lo) * SLDA + mloc);
        acc0 = wmma_f32_16x16x4(a, b0, acc0);
        acc1 = wmma_f32_16x16x4(a, b1, acc1);
      }
    }
    __syncthreads();
  }

  // ---- normalize and park att tile in LDS ----
  {
    const float i0 = 1.0f / sRowSum[n2a * 16 + lo];
    const float i1 = 1.0f / sRowSum[n2b * 16 + lo];
#pragma unroll
    for (int r = 0; r < 8; ++r) {
      sAtt[(c_off + r + 8 * hi) * 64 + n2a * 16 + lo] = acc0[r] * i0;
      sAtt[(c_off + r + 8 * hi) * 64 + n2b * 16 + lo] = acc1[r] * i1;
    }
  }
  __syncthreads();

  // ---- GEMM3: y = W2 * att, then BN + ReLU + residual ----
  const int o_off = (wave & 3) * 16;
#pragma unroll
  for (int i = 0; i < 2; ++i) {
    const int noff = ((wave >> 2) * 2 + i) * 16;
    v8f y = {};
#pragma unroll
    for (int kk = 0; kk < 16; ++kk) {
      const int c0 = kk * 4 + 2 * hi;
      v2f a;
      a.x = W2[(o_off + lo) * 64 + c0];
      a.y = W2[(o_off + lo) * 64 + c0 + 1];
      v2f bb;
      bb.x = sAtt[c0 * 64 + noff + lo];
      bb.y = sAtt[(c0 + 1) * 64 + noff + lo];
      y = wmma_f32_16x16x4(a, bb, y);
    }
#pragma unroll
    for (int r = 0; r < 8; ++r) {
      const int o = o_off + r + 8 * hi;
      const size_t idx = base + (size_t)o * N_DIM + nblk + noff + lo;
      out[idx] = fmaxf(y[r] * sInv[o] + sBias[o], 0.0f) + x[idx];
    }
  }
}

// ---------------------------------------------------------------------------
extern "C" void kernel_launch(void* const* d_in, const int* in_sizes, int n_in,
                              void* d_out, int out_size, void* d_ws,
                              size_t ws_size, hipStream_t stream) {
  const float* x     = (const float*)d_in[0];
  const float* Wk    = (const float*)d_in[1];
  const float* Wq    = (const float*)d_in[2];
  const float* Wv    = (const float*)d_in[3];
  const float* W2    = (const float*)d_in[4];
  const float* gamma = (const float*)d_in[5];
  const float* beta  = (const float*)d_in[6];
  const float* rmean = (const float*)d_in[7];
  const float* rvar  = (const float*)d_in[8];
  float* out = (float*)d_out;

  const size_t mat = (size_t)B_DIM * C_DIM * N_DIM;  // 1 Mi floats each
  float* kbuf = (float*)d_ws;
  float* qbuf = kbuf + mat;
  float* vbuf = qbuf + mat;

  qkv_proj_kernel<<<dim3(N_DIM / 128, B_DIM), 256, 0, stream>>>(
      x, Wk, Wq, Wv, kbuf, qbuf, vbuf);
  attn_fused_kernel<<<dim3(N_DIM / 64, B_DIM), 256, 0, stream>>>(
      kbuf, qbuf, vbuf, W2, gamma, beta, rmean, rvar, x, out);
}